// VDMGP_15496242004611
// MI455X (gfx1250) — compile-verified
//
#include <hip/hip_runtime.h>
#include <hip/hip_bf16.h>
#include <math.h>

typedef __attribute__((ext_vector_type(2))) float v2f;
typedef __attribute__((ext_vector_type(8))) float v8f;

#define NN 1024
#define DD 256
#define KK 8
#define MM 64
#define JITTER_F 1e-6f

// ---------------------------------------------------------------------------
// Kernel 1: per-n precompute.
//   qX[k,n]   = sum_d q_mu[k,d] X[n,d]
//   s[k,n]    = sum_d q_cov[k,d] X[n,d]^2
//   bot2      = 2 s + 1
//   Amat[n,k]     = 2 qX / bot2          (k < 8)
//   Amat[n,8+k]   = -1 / bot2
//   a2[n]     = sum_k ( -qX^2/bot2 - 0.5 log bot2 )
// ---------------------------------------------------------------------------
__global__ __launch_bounds__(256) void prep_n(const float* __restrict__ X,
                                              const float* __restrict__ q_mu,
                                              const float* __restrict__ q_cov,
                                              float* __restrict__ qX,
                                              float* __restrict__ sarr,
                                              float* __restrict__ Amat,
                                              float* __restrict__ a2) {
    int n = blockIdx.x * blockDim.x + threadIdx.x;
    if (n >= NN) return;
    float qx[KK], s[KK];
#pragma unroll
    for (int k = 0; k < KK; ++k) { qx[k] = 0.f; s[k] = 0.f; }
    const float* xr = X + (size_t)n * DD;
    for (int d = 0; d < DD; ++d) {
        float x = xr[d];
        float x2 = x * x;
#pragma unroll
        for (int k = 0; k < KK; ++k) {
            qx[k] = fmaf(q_mu[k * DD + d], x, qx[k]);
            s[k]  = fmaf(q_cov[k * DD + d], x2, s[k]);
        }
    }
    float a2v = 0.f;
#pragma unroll
    for (int k = 0; k < KK; ++k) {
        qX[k * NN + n]   = qx[k];
        sarr[k * NN + n] = s[k];
        float bot2 = 2.f * s[k] + 1.f;
        float inv  = 1.f / bot2;
        Amat[n * 16 + k]     = 2.f * qx[k] * inv;
        Amat[n * 16 + 8 + k] = -inv;
        a2v += -qx[k] * qx[k] * inv - 0.5f * logf(bot2);
    }
    a2[n] = a2v;
}

// ---------------------------------------------------------------------------
// Kernel 2: per-(i,j) precompute over Z.
//   Bmat[ij,k]   = Zbar[k]        (k < 8)
//   Bmat[ij,8+k] = Zbar[k]^2
//   leftM[ij]    = sigma_f^2 exp(-0.25 sum_k (Zi-Zj)^2)
//   Kuu[ij]      = kern_var exp(-0.5 sum_k ((Zi-Zj)/ls)^2) + JITTER*(i==j)
// ---------------------------------------------------------------------------
__global__ __launch_bounds__(256) void prep_z(const float* __restrict__ Z,
                                              const float* __restrict__ sigma_f,
                                              const float* __restrict__ kern_var,
                                              const float* __restrict__ kern_ls,
                                              float* __restrict__ Bmat,
                                              float* __restrict__ leftM,
                                              float* __restrict__ Kuu) {
    int ij = blockIdx.x * blockDim.x + threadIdx.x;
    if (ij >= MM * MM) return;
    int i = ij >> 6, j = ij & 63;
    float d2 = 0.f, d2ls = 0.f;
#pragma unroll
    for (int k = 0; k < KK; ++k) {
        float zi = Z[i * KK + k];
        float zj = Z[j * KK + k];
        float zb = 0.5f * (zi + zj);
        Bmat[ij * 16 + k]     = zb;
        Bmat[ij * 16 + 8 + k] = zb * zb;
        float dd = zi - zj;
        d2 += dd * dd;
        float ddl = dd / kern_ls[k];
        d2ls += ddl * ddl;
    }
    float sf = sigma_f[0];
    leftM[ij] = sf * sf * expf(-0.25f * d2);
    Kuu[ij]   = kern_var[0] * expf(-0.5f * d2ls) + ((i == j) ? JITTER_F : 0.f);
}

// ---------------------------------------------------------------------------
// Kernel 3: Psi1_Y[m] = sum_n psi1[n,m] * Y[n]  (deterministic LDS reduction)
//   log psi1[n,m] = log sf + sum_k( -0.5 (qX - Z[m,k])^2 / bot1 - 0.5 log bot1 )
// ---------------------------------------------------------------------------
__global__ __launch_bounds__(256) void psi1y_kernel(const float* __restrict__ qX,
                                                    const float* __restrict__ sarr,
                                                    const float* __restrict__ Y,
                                                    const float* __restrict__ Z,
                                                    const float* __restrict__ sigma_f,
                                                    float* __restrict__ psi1y) {
    int m = blockIdx.x;      // 0..63
    int t = threadIdx.x;     // 0..255
    float zm[KK];
#pragma unroll
    for (int k = 0; k < KK; ++k) zm[k] = Z[m * KK + k];
    float lsf = logf(sigma_f[0]);
    float acc = 0.f;
    for (int n = t; n < NN; n += 256) {
        float e = lsf;
#pragma unroll
        for (int k = 0; k < KK; ++k) {
            float qx   = qX[k * NN + n];
            float bot1 = 1.f + sarr[k * NN + n];
            float diff = qx - zm[k];
            e += -0.5f * diff * diff / bot1 - 0.5f * logf(bot1);
        }
        acc += expf(e) * Y[n];
    }
    __shared__ float red[256];
    red[t] = acc;
    __syncthreads();
    for (int sft = 128; sft > 0; sft >>= 1) {
        if (t < sft) red[t] += red[t + sft];
        __syncthreads();
    }
    if (t == 0) psi1y[m] = red[0];
}

// ---------------------------------------------------------------------------
// Kernel 4: psi2 via WMMA f32 16x16x4 GEMM in log space.
//   logright[n,ij] = a2[n] + Amat[n,:] . Bmat[ij,:]   (K = 16)
//   psi2[ij] = leftM[ij] * sum_n exp(logright[n,ij])
// One wave per 16 output columns; B-fragments stay resident.
// ---------------------------------------------------------------------------
__global__ __launch_bounds__(32) void psi2_wmma(const float* __restrict__ Amat,
                                                const float* __restrict__ a2,
                                                const float* __restrict__ Bmat,
                                                const float* __restrict__ leftM,
                                                float* __restrict__ psi2) {
    const int lane = threadIdx.x;       // 0..31
    const int half = lane >> 4;         // 0 | 1
    const int l15  = lane & 15;
    const int j0   = blockIdx.x * 16;
    const int col  = j0 + l15;
    const int mo   = half * 8;          // C-layout M offset for this lane half

    // B fragment: 16x(K=16) columns, 4 K-chunks of 4.
    // f32 16x16x4 layout: lanes 0-15 carry K pair {0,1}, lanes 16-31 carry {2,3}.
    v2f bfrag[4];
#pragma unroll
    for (int kc = 0; kc < 4; ++kc) {
        const float* bp = &Bmat[col * 16 + kc * 4 + half * 2];
        bfrag[kc].x = bp[0];
        bfrag[kc].y = bp[1];
    }

    float acc[8];
#pragma unroll
    for (int r = 0; r < 8; ++r) acc[r] = 0.f;

    for (int n0 = 0; n0 < NN; n0 += 16) {
        v8f c = {0.f, 0.f, 0.f, 0.f, 0.f, 0.f, 0.f, 0.f};
#pragma unroll
        for (int kc = 0; kc < 4; ++kc) {
            const float* ap = &Amat[(n0 + l15) * 16 + kc * 4 + half * 2];
            v2f a;
            a.x = ap[0];
            a.y = ap[1];
            // 8 args: (neg_a, A, neg_b, B, c_mod, C, reuse_a, reuse_b)
            c = __builtin_amdgcn_wmma_f32_16x16x4_f32(false, a, false, bfrag[kc],
                                                      (short)0, c, false, false);
        }
        // C element (VGPR r, this lane) is row n = n0 + mo + r, col = col.
#pragma unroll
        for (int r = 0; r < 8; ++r) {
            acc[r] += __expf(c[r] + a2[n0 + mo + r]);
        }
    }
    float t = 0.f;
#pragma unroll
    for (int r = 0; r < 8; ++r) t += acc[r];
    t += __shfl_xor(t, 16, 32);          // merge the two lane halves (same col)
    if (half == 0) psi2[col] = leftM[col] * t;
}

// ---------------------------------------------------------------------------
// Kernel 5: finalize — Choleskys, triangular solves, reductions, scalar out.
// Single block of 64 threads.
// ---------------------------------------------------------------------------
__device__ void chol64(float* A, int t) {
    for (int k = 0; k < MM; ++k) {
        if (t == 0) A[k * MM + k] = sqrtf(A[k * MM + k]);
        __syncthreads();
        if (t > k) A[t * MM + k] /= A[k * MM + k];
        __syncthreads();
        if (t > k) {
            float lrk = A[t * MM + k];
            for (int c = k + 1; c <= t; ++c) A[t * MM + c] -= lrk * A[c * MM + k];
        }
        __syncthreads();
    }
}

__device__ float reduce64(float* red, int t, float v) {
    red[t] = v;
    __syncthreads();
    for (int s = 32; s > 0; s >>= 1) {
        if (t < s) red[t] += red[t + s];
        __syncthreads();
    }
    float r = red[0];
    __syncthreads();
    return r;
}

__global__ __launch_bounds__(64) void finalize(const float* __restrict__ Kuu_g,
                                               const float* __restrict__ psi2_g,
                                               const float* __restrict__ psi1y_g,
                                               const float* __restrict__ Y,
                                               const float* __restrict__ q_mu,
                                               const float* __restrict__ q_cov,
                                               const float* __restrict__ sigma_f,
                                               const float* __restrict__ lik_var,
                                               float* __restrict__ out) {
    __shared__ float LKu[MM * MM];
    __shared__ float LKP[MM * MM];
    __shared__ float xv[MM];
    __shared__ float red[64];
    __shared__ float klv[KK];
    int t = threadIdx.x;
    float sigma2 = lik_var[0];

    for (int idx = t; idx < MM * MM; idx += 64) {
        float ku = Kuu_g[idx];
        LKu[idx] = ku;
        int i = idx >> 6, j = idx & 63;
        LKP[idx] = sigma2 * ku + psi2_g[idx] + ((i == j) ? JITTER_F : 0.f);
    }
    __syncthreads();

    chol64(LKu, t);
    chol64(LKP, t);

    float sumlogKu = reduce64(red, t, logf(LKu[t * MM + t]));
    float sumlogKP = reduce64(red, t, logf(LKP[t * MM + t]));

    // forward solve:  LKP * x = psi1y
    xv[t] = psi1y_g[t];
    __syncthreads();
    for (int r = 0; r < MM; ++r) {
        if (t == r) xv[r] /= LKP[r * MM + r];
        __syncthreads();
        if (t > r) xv[t] -= LKP[t * MM + r] * xv[r];
        __syncthreads();
    }
    float dotPsol = reduce64(red, t, psi1y_g[t] * xv[t]);

    // trace(LKu^{-1} psi2): thread t solves column t, keeps diagonal entry.
    float xc[MM];
    float trj = 0.f;
    for (int r = 0; r < MM; ++r) {
        float v = psi2_g[r * MM + t];
        for (int c = 0; c < r; ++c) v -= LKu[r * MM + c] * xc[c];
        v /= LKu[r * MM + r];
        xc[r] = v;
        if (r == t) trj = v;
    }
    float trace = reduce64(red, t, trj);

    // YY = Y^T Y
    float yy = 0.f;
    for (int n = t; n < NN; n += 64) { float y = Y[n]; yy += y * y; }
    float YY = reduce64(red, t, yy);

    // KL
    if (t < KK) {
        float sl = 0.f, ss = 0.f;
        for (int d = 0; d < DD; ++d) {
            float qc = q_cov[t * DD + d];
            float qm = q_mu[t * DD + d];
            sl += logf(qc);
            ss += qc + qm * qm;
        }
        klv[t] = sl - (float)DD * logf(ss) + (float)DD * logf((float)DD);
    }
    __syncthreads();

    if (t == 0) {
        float KL = 0.f;
        for (int k = 0; k < KK; ++k) KL += klv[k];
        float sf   = sigma_f[0];
        float psi0 = (float)NN * sf * sf;
        const float LOG2PI = 1.8378770664093453f;
        float F1 = -(float)NN * LOG2PI - (float)(NN - MM) * logf(sigma2) + sumlogKu;
        F1 -= sumlogKP;
        F1 -= YY / sigma2;
        F1 += dotPsol / sigma2;
        F1 += -psi0 / sigma2 + trace / sigma2;
        out[0] = 0.5f * (F1 + KL);
    }
}

// ---------------------------------------------------------------------------
extern "C" void kernel_launch(void* const* d_in, const int* in_sizes, int n_in,
                              void* d_out, int out_size, void* d_ws, size_t ws_size,
                              hipStream_t stream) {
    const float* X        = (const float*)d_in[0];
    const float* Y        = (const float*)d_in[1];
    const float* q_mu     = (const float*)d_in[2];
    const float* q_cov    = (const float*)d_in[3];
    const float* Z        = (const float*)d_in[4];
    const float* sigma_f  = (const float*)d_in[5];
    const float* lik_var  = (const float*)d_in[6];
    const float* kern_var = (const float*)d_in[7];
    const float* kern_ls  = (const float*)d_in[8];

    float* ws    = (float*)d_ws;
    float* qX    = ws;                    // 8*1024
    float* sarr  = qX    + KK * NN;       // 8*1024
    float* Amat  = sarr  + KK * NN;       // 1024*16
    float* a2    = Amat  + NN * 16;       // 1024
    float* Bmat  = a2    + NN;            // 4096*16
    float* leftM = Bmat  + MM * MM * 16;  // 4096
    float* Kuu   = leftM + MM * MM;       // 4096
    float* psi2  = Kuu   + MM * MM;       // 4096
    float* psi1y = psi2  + MM * MM;       // 64

    prep_n<<<NN / 256, 256, 0, stream>>>(X, q_mu, q_cov, qX, sarr, Amat, a2);
    prep_z<<<(MM * MM) / 256, 256, 0, stream>>>(Z, sigma_f, kern_var, kern_ls,
                                                Bmat, leftM, Kuu);
    psi1y_kernel<<<MM, 256, 0, stream>>>(qX, sarr, Y, Z, sigma_f, psi1y);
    psi2_wmma<<<(MM * MM) / 16, 32, 0, stream>>>(Amat, a2, Bmat, leftM, psi2);
    finalize<<<1, 64, 0, stream>>>(Kuu, psi2, psi1y, Y, q_mu, q_cov,
                                   sigma_f, lik_var, (float*)d_out);
}